// GuidedBlurLayer_72550587564721
// MI455X (gfx1250) — compile-verified
//
#include <hip/hip_runtime.h>

// ---------------------------------------------------------------------------
// Bilateral blur, 16x3x512x512 f32, KS=9, sigma_color=0.1, sigma_y=0.1, sigma_x=2.0
//
// f32-exact reduction: gaussian_1d(9, 0.1) in float32 is
//   [0,0,0, 1.9287e-22, 1.0, 1.9287e-22, 0,0,0]   (exp(-200) underflows to 0.0f)
// => only tap rows dy in {-1,0,+1} have nonzero spatial weight; the other 6
//    rows contribute exactly 0.0f to num and den. 27 taps instead of 81.
//
// Inner loop folds spatial weight + range kernel into ONE v_exp_f32 per tap:
//   w = space * exp(-50*d^2) = exp2( d^2 * (-50/ln2) + log2(space) )
// ---------------------------------------------------------------------------

#define IMG_H 512
#define IMG_W 512
#define TS_X  64          // output tile width  per block
#define TS_Y  16          // output tile height per block
#define HALO_X 4
#define HALO_Y 1          // only dy in {-1,0,1} has nonzero f32 spatial weight
#define LDS_W (TS_X + 2*HALO_X)   // 72 floats per row (288B, 16B multiple)
#define LDS_H (TS_Y + 2*HALO_Y)   // 18 rows

#define NEG50_OVER_LN2 (-72.134752f)   // -50 / ln(2)

#if __has_builtin(__builtin_amdgcn_global_load_async_to_lds_b32)
#define HAVE_ASYNC_LDS 1
typedef __attribute__((address_space(1))) int* gptr_t;
typedef __attribute__((address_space(3))) int* sptr_t;
#endif

__device__ __forceinline__ float fast_exp2(float x) {
#if __has_builtin(__builtin_amdgcn_exp2f)
    return __builtin_amdgcn_exp2f(x);     // raw v_exp_f32
#else
    return exp2f(x);
#endif
}

__device__ __forceinline__ int reflect512(int p) {
    p = (p < 0) ? -p : p;
    p = (p > 511) ? (1022 - p) : p;
    return p;
}

__global__ __launch_bounds__(256)
void bilateral_rows3_kernel(const float* __restrict__ in, float* __restrict__ out) {
    __shared__ __align__(16) float sh[3][LDS_H][LDS_W];

    const int tid = threadIdx.x;
    const int b   = blockIdx.z;
    const int by  = blockIdx.y * TS_Y;
    const int bx  = blockIdx.x * TS_X;

    // ---------------- stage halo tile (3 channel planes) into LDS ----------
    const int NTOT = 3 * LDS_H * LDS_W;   // 3888 elements
    for (int f = tid; f < NTOT; f += 256) {
        const int xx = f % LDS_W;
        const int r  = f / LDS_W;
        const int yy = r % LDS_H;
        const int c  = r / LDS_H;
        const int gy = reflect512(by + yy - HALO_Y);
        const int gx = reflect512(bx + xx - HALO_X);
        const float* gsrc = in + (((size_t)(b * 3 + c) * IMG_H + gy) * IMG_W + gx);
#ifdef HAVE_ASYNC_LDS
        // CDNA5 async global->LDS copy (ASYNCcnt path); per-lane gather.
        __builtin_amdgcn_global_load_async_to_lds_b32(
            (gptr_t)const_cast<float*>(gsrc),
            (sptr_t)&sh[c][yy][xx],
            0, 0);
#else
        sh[c][yy][xx] = *gsrc;
#endif
    }
#ifdef HAVE_ASYNC_LDS
#if __has_builtin(__builtin_amdgcn_s_wait_asynccnt)
    __builtin_amdgcn_s_wait_asynccnt(0);
#else
    asm volatile("s_wait_asynccnt 0" ::: "memory");
#endif
#endif
    __syncthreads();

    // ------- per-thread spatial weights in log2 domain (one-time cost) -----
    float gx9[9];
    float s = 0.0f;
#pragma unroll
    for (int j = 0; j < 9; ++j) {
        const float t = (float)(j - 4);
        gx9[j] = __expf(-0.125f * t * t);   // exp(-0.5*(t/2)^2), f32 as reference
        s += gx9[j];
    }
    const float invs = 1.0f / s;
    float lgx9[9];
#pragma unroll
    for (int j = 0; j < 9; ++j) lgx9[j] = __log2f(gx9[j] * invs);
    // rows dy=+-1: spatial weight exp(-50) ~ 1.93e-22  =>  log2 = -50/ln2
    const float lgy_off = NEG50_OVER_LN2;

    // ---------------- compute: 4 pixels per thread along x -----------------
    const int tx = (tid & 15) * 4;          // 0..60, 16B aligned in LDS row
    const int ty = tid >> 4;                // 0..15

    float cen[3][4];
#pragma unroll
    for (int c = 0; c < 3; ++c)
#pragma unroll
        for (int p = 0; p < 4; ++p)
            cen[c][p] = sh[c][ty + HALO_Y][tx + HALO_X + p];

    float num[3][4] = {{0.f}};
    float den[4]    = {0.f};

#pragma unroll
    for (int i = 0; i < 3; ++i) {           // dy = -1, 0, +1
        const float lrow = (i == 1) ? 0.0f : lgy_off;

        // 12 contiguous neighbor columns per channel: three aligned b128 loads
        float row[3][12];
#pragma unroll
        for (int c = 0; c < 3; ++c) {
            const float4 a = *(const float4*)&sh[c][ty + i][tx + 0];
            const float4 m = *(const float4*)&sh[c][ty + i][tx + 4];
            const float4 z = *(const float4*)&sh[c][ty + i][tx + 8];
            row[c][0] = a.x;  row[c][1] = a.y;  row[c][2]  = a.z;  row[c][3]  = a.w;
            row[c][4] = m.x;  row[c][5] = m.y;  row[c][6]  = m.z;  row[c][7]  = m.w;
            row[c][8] = z.x;  row[c][9] = z.y;  row[c][10] = z.z;  row[c][11] = z.w;
        }

#pragma unroll
        for (int j = 0; j < 9; ++j) {       // dx = -4..+4
            const float lw = lrow + lgx9[j];          // log2(space weight)
#pragma unroll
            for (int p = 0; p < 4; ++p) {
                const float n0 = row[0][p + j];
                const float n1 = row[1][p + j];
                const float n2 = row[2][p + j];
                const float d  = fabsf(n0 - cen[0][p]) +
                                 fabsf(n1 - cen[1][p]) +
                                 fabsf(n2 - cen[2][p]);
                // w = space * exp(-50*d^2) in a single v_exp_f32
                const float w  = fast_exp2(fmaf(d * d, NEG50_OVER_LN2, lw));
                num[0][p] += n0 * w;
                num[1][p] += n1 * w;
                num[2][p] += n2 * w;
                den[p]    += w;
            }
        }
    }

    // ---------------- normalize, clip, store (aligned float4) --------------
    float rden[4];
#pragma unroll
    for (int p = 0; p < 4; ++p) rden[p] = 1.0f / den[p];

    const int gy  = by + ty;
    const int gx0 = bx + tx;
#pragma unroll
    for (int c = 0; c < 3; ++c) {
        float4 o;
        o.x = fminf(fmaxf(num[c][0] * rden[0], 0.0f), 1.0f);
        o.y = fminf(fmaxf(num[c][1] * rden[1], 0.0f), 1.0f);
        o.z = fminf(fmaxf(num[c][2] * rden[2], 0.0f), 1.0f);
        o.w = fminf(fmaxf(num[c][3] * rden[3], 0.0f), 1.0f);
        *(float4*)&out[(((size_t)(b * 3 + c) * IMG_H + gy) * IMG_W + gx0)] = o;
    }
}

extern "C" void kernel_launch(void* const* d_in, const int* in_sizes, int n_in,
                              void* d_out, int out_size, void* d_ws, size_t ws_size,
                              hipStream_t stream) {
    const float* x   = (const float*)d_in[0];
    float*       out = (float*)d_out;
    dim3 grid(IMG_W / TS_X, IMG_H / TS_Y, 16);   // (8, 32, 16)
    bilateral_rows3_kernel<<<grid, dim3(256), 0, stream>>>(x, out);
}